// myMSEloss_35914516529866
// MI455X (gfx1250) — compile-verified
//
#include <hip/hip_runtime.h>

typedef float v2f __attribute__((ext_vector_type(2)));
typedef float v4f __attribute__((ext_vector_type(4)));
typedef float v8f __attribute__((ext_vector_type(8)));

__device__ __forceinline__ float sqf(float x) { return x * x; }

// Masked row loss: COUNTS[c] = 3c+1, so groups {1..3},{4..6},{7..9} gate on c>=1,2,3.
__device__ __forceinline__ float row_loss(const float o[10], const float l[10]) {
    float c  = l[0];  // class id stored as float 0..3
    float g0 = sqf(o[0] - l[0]);
    float g1 = sqf(o[1] - l[1]) + sqf(o[2] - l[2]) + sqf(o[3] - l[3]);
    float g2 = sqf(o[4] - l[4]) + sqf(o[5] - l[5]) + sqf(o[6] - l[6]);
    float g3 = sqf(o[7] - l[7]) + sqf(o[8] - l[8]) + sqf(o[9] - l[9]);
    float s = g0;
    s += (c >= 1.0f) ? g1 : 0.0f;
    s += (c >= 2.0f) ? g2 : 0.0f;
    s += (c >= 3.0f) ? g3 : 0.0f;
    return s;
}

// Block-wide reduction of 256 per-thread partials using v_wmma_f32_16x16x4_f32.
// A holds 64 partials (16x4 f32, 2 VGPRs/lane), B is all-ones -> every D row m
// equals sum_k A[m,k]; 4 accumulating WMMAs cover all 256 values. Column n of D
// lives in lane n (rows 0-7) and lane n+16 (rows 8-15): fold with one shuffle.
// Result valid in lane 0 of wave 0 (threadIdx.x == 0).
__device__ __forceinline__ float block_reduce_wmma(float p, float* lds) {
    const int tid = threadIdx.x;
    lds[tid] = p;
    __syncthreads();
    float total = 0.0f;
    if (tid < 32) {                 // whole wave 0 -> EXEC all ones (WMMA requirement)
        v8f acc = {};               // C = 0
        v2f bones;
        bones.x = 1.0f;
        bones.y = 1.0f;
#pragma unroll
        for (int i = 0; i < 4; ++i) {
            v2f a;
            a.x = lds[i * 64 + tid];
            a.y = lds[i * 64 + 32 + tid];
            acc = __builtin_amdgcn_wmma_f32_16x16x4_f32(
                /*neg_a=*/false, a, /*neg_b=*/false, bones,
                /*c_mod=*/(short)0, acc, /*reuse_a=*/false, /*reuse_b=*/false);
        }
        float s = acc[0] + acc[1] + acc[2] + acc[3] +
                  acc[4] + acc[5] + acc[6] + acc[7];
        s += __shfl_down(s, 16, 32);  // lane n + lane n+16 = full column sum
        total = s;                    // lane 0 holds the block total
    }
    return total;
}

// Pass 1: stream row pairs (80B aligned chunks -> 5x b128 NT loads per array),
// accumulate masked SSD, write one partial per block.
__global__ __launch_bounds__(256) void mse_partial_kernel(
    const float* __restrict__ outs, const float* __restrict__ labs,
    float* __restrict__ partials, long long nrows) {
    __shared__ float lds[256];

    const long long npair   = nrows >> 1;
    const long long gtid    = (long long)blockIdx.x * blockDim.x + threadIdx.x;
    const long long gstride = (long long)gridDim.x * blockDim.x;

    float acc = 0.0f;
    for (long long pr = gtid; pr < npair; pr += gstride) {
        const v4f* po = (const v4f*)(outs + pr * 20);  // 80B aligned
        const v4f* pl = (const v4f*)(labs + pr * 20);
        v4f o0 = __builtin_nontemporal_load(po + 0);
        v4f o1 = __builtin_nontemporal_load(po + 1);
        v4f o2 = __builtin_nontemporal_load(po + 2);
        v4f o3 = __builtin_nontemporal_load(po + 3);
        v4f o4 = __builtin_nontemporal_load(po + 4);
        v4f l0 = __builtin_nontemporal_load(pl + 0);
        v4f l1 = __builtin_nontemporal_load(pl + 1);
        v4f l2 = __builtin_nontemporal_load(pl + 2);
        v4f l3 = __builtin_nontemporal_load(pl + 3);
        v4f l4 = __builtin_nontemporal_load(pl + 4);

        const float oa[10] = {o0.x, o0.y, o0.z, o0.w, o1.x, o1.y, o1.z, o1.w, o2.x, o2.y};
        const float la[10] = {l0.x, l0.y, l0.z, l0.w, l1.x, l1.y, l1.z, l1.w, l2.x, l2.y};
        const float ob[10] = {o2.z, o2.w, o3.x, o3.y, o3.z, o3.w, o4.x, o4.y, o4.z, o4.w};
        const float lb[10] = {l2.z, l2.w, l3.x, l3.y, l3.z, l3.w, l4.x, l4.y, l4.z, l4.w};

        acc += row_loss(oa, la);
        acc += row_loss(ob, lb);
    }

    // Odd-N tail row (N=4M is even; kept for generality), handled once globally.
    if ((nrows & 1) && gtid == 0) {
        const float* o = outs + (nrows - 1) * 10;
        const float* l = labs + (nrows - 1) * 10;
        float ot[10], lt[10];
#pragma unroll
        for (int j = 0; j < 10; ++j) { ot[j] = o[j]; lt[j] = l[j]; }
        acc += row_loss(ot, lt);
    }

    float bsum = block_reduce_wmma(acc, lds);
    if (threadIdx.x == 0) partials[blockIdx.x] = bsum;
}

// Pass 2: one block folds all block partials, scales by 1/N, writes scalar.
__global__ __launch_bounds__(256) void mse_final_kernel(
    const float* __restrict__ partials, int nparts,
    float* __restrict__ out, float inv_n) {
    __shared__ float lds[256];
    float acc = 0.0f;
    for (int i = threadIdx.x; i < nparts; i += 256) acc += partials[i];
    float total = block_reduce_wmma(acc, lds);
    if (threadIdx.x == 0) out[0] = total * inv_n;
}

extern "C" void kernel_launch(void* const* d_in, const int* in_sizes, int n_in,
                              void* d_out, int out_size, void* d_ws, size_t ws_size,
                              hipStream_t stream) {
    const float* outs = (const float*)d_in[0];
    const float* labs = (const float*)d_in[1];
    const long long nelem = (long long)in_sizes[0];
    const long long nrows = nelem / 10;   // D = 10

    float* partials = (float*)d_ws;
    int nblocks = 2048;                   // ~4 row-pairs per thread at N=4M
    size_t maxb = ws_size / sizeof(float);
    if ((size_t)nblocks > maxb) nblocks = (int)maxb;
    if (nblocks < 1) nblocks = 1;

    mse_partial_kernel<<<nblocks, 256, 0, stream>>>(outs, labs, partials, nrows);
    mse_final_kernel<<<1, 256, 0, stream>>>(partials, nblocks, (float*)d_out,
                                            1.0f / (float)nrows);
}